// DifferentiableChebyshevOperator_65687229825052
// MI455X (gfx1250) — compile-verified
//
#include <hip/hip_runtime.h>
#include <cmath>
#include <cstdint>
#include <cstddef>

// ---------------------------------------------------------------------------
// Chebyshev heat-kernel filter: acc = sum_k c_k T_k(L_rescaled) X
//   N=100000 nodes, d=64 features, nnz=1.7M COO entries (incl. self loops).
//   Device-built packed CSR (once), then 29 fused SpMM+recurrence kernels.
//   Working set (~90MB) is L2-resident on MI455X (192MB L2) -> L2-gather bound.
//   CDNA5 paths: global_load_async_to_lds_b64 staging of the (col,val) stream
//   (ASYNCcnt), s_wait_asynccnt, global_prefetch, wave32 256B-burst gathers.
// ---------------------------------------------------------------------------

#define CHEB_M   30
#define WGROWS   16         // rows per workgroup (2 rows per wave32)
#define LDS_CAP  4096       // (col,val) pairs staged in LDS per workgroup (32KB)

// ---------------------------- CSR build ------------------------------------

__global__ void zero_kernel(int* __restrict__ row_start, int* __restrict__ cursor, int N) {
  int i = blockIdx.x * blockDim.x + threadIdx.x;
  if (i <= N) row_start[i] = 0;
  if (i < N)  cursor[i] = 0;
}

__global__ void count_kernel(const int* __restrict__ rows, int* __restrict__ row_start, int nnz) {
  int j = blockIdx.x * blockDim.x + threadIdx.x;
  if (j < nnz) atomicAdd(&row_start[rows[j]], 1);
}

// Single-workgroup exclusive scan over N counts (runs once; ~98 chunk
// iterations of a 1024-wide Hillis-Steele scan, negligible vs 29 SpMMs).
__global__ void scan_kernel(int* __restrict__ data, int n) {
  __shared__ int sm[1024];
  __shared__ int carry_sm;
  const int tid = threadIdx.x;
  if (tid == 0) carry_sm = 0;
  __syncthreads();
  for (int base = 0; base < n; base += 1024) {
    int i = base + tid;
    int v = (i < n) ? data[i] : 0;
    sm[tid] = v;
    __syncthreads();
    for (int off = 1; off < 1024; off <<= 1) {
      int t = (tid >= off) ? sm[tid - off] : 0;
      __syncthreads();
      sm[tid] += t;
      __syncthreads();
    }
    int incl = sm[tid];
    int c = carry_sm;
    if (i < n) data[i] = c + incl - v;   // exclusive prefix
    __syncthreads();
    if (tid == 1023) carry_sm = c + incl;
    __syncthreads();
  }
  if (tid == 0) data[n] = carry_sm;
}

__global__ void fill_kernel(const int* __restrict__ rows, const int* __restrict__ cols,
                            const float* __restrict__ vals, const int* __restrict__ row_start,
                            int* __restrict__ cursor, int2* __restrict__ pairs, int nnz) {
  int j = blockIdx.x * blockDim.x + threadIdx.x;
  if (j < nnz) {
    int r = rows[j];
    int pos = atomicAdd(&cursor[r], 1);
    pairs[row_start[r] + pos] = make_int2(cols[j], __float_as_int(vals[j]));
  }
}

// ------------------------ fused SpMM + recurrence ---------------------------
// MODE 0 (init):  z = L@X ;  T1 = z - X ;         dst=T1 ; acc  = c0*X + c1*T1
// MODE 1 (step):  z = L@T1;  Tk = 2z - 2*T1 - T0;  dst=Tk ; acc += ck*Tk
// Layout: 2 rows per wave32; each half-wave (16 lanes) owns one row with a
// float4 of features per lane -> every neighbor gather is one coalesced 256B
// global_load_b128 burst per half-wave, with 4 independent FMA chains/lane
// and two independent rows per wave for latency hiding against L2.
template <int MODE>
__global__ __launch_bounds__(256)
void cheb_spmm_step(const int2* __restrict__ pairs,
                    const int*  __restrict__ row_start,
                    const float* __restrict__ Yin,   // T_{k-1} (never aliases dst)
                    const float* T0,                 // T_{k-2} (may alias Tdst)
                    float* Tdst,                     // T_k
                    float* __restrict__ acc,
                    float cA, float cB, int N) {
  __shared__ int2 spairs[LDS_CAP];
  const int r0 = blockIdx.x * WGROWS;
  if (r0 >= N) return;
  const int rend = (r0 + WGROWS < N) ? (r0 + WGROWS) : N;
  const int base = row_start[r0];
  const int cnt  = row_start[rend] - base;
  const bool useLds = (cnt <= LDS_CAP);

  if (useLds) {
    // Cold-pass L2 prefetch of the next chunk of the pair stream.
    if (threadIdx.x == 0) __builtin_prefetch(pairs + base + cnt, 0, 1);
    for (int j = threadIdx.x; j < cnt; j += blockDim.x) {
      unsigned lds_addr = (unsigned)(size_t)(&spairs[j]);   // low 32 bits = LDS addr
      const int2* gp = pairs + base + j;
      asm volatile("global_load_async_to_lds_b64 %0, %1, off"
                   :: "v"(lds_addr), "v"(gp) : "memory");
    }
    asm volatile("s_wait_asynccnt 0x0" ::: "memory");
  }
  __syncthreads();

  const int wid  = threadIdx.x >> 5;          // wave id: 0..7
  const int half = (threadIdx.x >> 4) & 1;    // half-wave -> row select
  const int fl   = threadIdx.x & 15;          // feature lane: float4 index
  const int row  = r0 + wid * 2 + half;
  if (row >= N) return;

  const int s = row_start[row];
  const int e = row_start[row + 1];
  const float4* __restrict__ Y4 = (const float4*)Yin;

  float4 z = {0.f, 0.f, 0.f, 0.f};
  if (useLds) {
#pragma unroll 2
    for (int j = s; j < e; ++j) {
      int2 p = spairs[j - base];                 // LDS broadcast per half-wave
      float v = __int_as_float(p.y);
      float4 y = Y4[(size_t)p.x * 16 + fl];      // coalesced 256B row gather
      z.x = fmaf(v, y.x, z.x);
      z.y = fmaf(v, y.y, z.y);
      z.z = fmaf(v, y.z, z.z);
      z.w = fmaf(v, y.w, z.w);
    }
  } else {
#pragma unroll 2
    for (int j = s; j < e; ++j) {
      int2 p = pairs[j];
      float v = __int_as_float(p.y);
      float4 y = Y4[(size_t)p.x * 16 + fl];
      z.x = fmaf(v, y.x, z.x);
      z.y = fmaf(v, y.y, z.y);
      z.z = fmaf(v, y.z, z.z);
      z.w = fmaf(v, y.w, z.w);
    }
  }

  const size_t off = (size_t)row * 16 + fl;
  float4* D4 = (float4*)Tdst;
  float4* A4 = (float4*)acc;
  if (MODE == 0) {
    const float4 x = ((const float4*)T0)[off];   // T0 == X here
    float4 t1v = {z.x - x.x, z.y - x.y, z.z - x.z, z.w - x.w};
    D4[off] = t1v;
    float4 a = {cA * x.x + cB * t1v.x, cA * x.y + cB * t1v.y,
                cA * x.z + cB * t1v.z, cA * x.w + cB * t1v.w};
    A4[off] = a;                                 // full write (no pre-zero needed)
  } else {
    const float4 t1v = Y4[off];
    const float4 t0v = ((const float4*)T0)[off]; // read BEFORE in-place overwrite
    float4 tk = {2.f * z.x - 2.f * t1v.x - t0v.x,
                 2.f * z.y - 2.f * t1v.y - t0v.y,
                 2.f * z.z - 2.f * t1v.z - t0v.z,
                 2.f * z.w - 2.f * t1v.w - t0v.w};
    D4[off] = tk;
    float4 a = A4[off];
    a.x = fmaf(cA, tk.x, a.x);
    a.y = fmaf(cA, tk.y, a.y);
    a.z = fmaf(cA, tk.z, a.z);
    a.w = fmaf(cA, tk.w, a.w);
    A4[off] = a;
  }
  (void)cB;
}

// ------------------------------- launcher -----------------------------------

extern "C" void kernel_launch(void* const* d_in, const int* in_sizes, int n_in,
                              void* d_out, int out_size, void* d_ws, size_t ws_size,
                              hipStream_t stream) {
  (void)n_in; (void)out_size; (void)ws_size;
  const int*   rows = (const int*)d_in[0];
  const int*   cols = (const int*)d_in[1];
  const float* vals = (const float*)d_in[2];
  const float* X    = (const float*)d_in[3];
  float* acc = (float*)d_out;

  const int nnz = in_sizes[0];
  const int N   = in_sizes[3] / 64;

  // Chebyshev coefficients of exp(-t*lambda), t=5, lambda_max=2 (host, double).
  const double PI = 3.14159265358979323846;
  double c[CHEB_M];
  for (int k = 0; k < CHEB_M; ++k) {
    double s = 0.0;
    for (int j = 0; j < CHEB_M; ++j) {
      double x = cos(PI * (j + 0.5) / (double)CHEB_M);
      s += exp(-5.0 * (x + 1.0)) * cos((double)k * acos(x));
    }
    c[k] = 2.0 / (double)CHEB_M * s;
  }
  c[0] *= 0.5;

  // Workspace carve-up (256B aligned): row_start | cursor | pairs | bufA | bufB
  char* p = (char*)d_ws;
  auto carve = [&p](size_t bytes) -> void* {
    void* r = (void*)p;
    p += (bytes + 255) & ~(size_t)255;
    return r;
  };
  int*  row_start = (int*)carve((size_t)(N + 1) * sizeof(int));
  int*  cursor    = (int*)carve((size_t)N * sizeof(int));
  int2* pairs     = (int2*)carve((size_t)nnz * sizeof(int2));
  float* bufA     = (float*)carve((size_t)N * 64 * sizeof(float));
  float* bufB     = (float*)carve((size_t)N * 64 * sizeof(float));

  // 1) CSR build (deterministic work each call; cheap vs 29 SpMMs).
  zero_kernel <<<(N + 256) / 256, 256, 0, stream>>>(row_start, cursor, N);
  count_kernel<<<(nnz + 255) / 256, 256, 0, stream>>>(rows, row_start, nnz);
  scan_kernel <<<1, 1024, 0, stream>>>(row_start, N);
  fill_kernel <<<(nnz + 255) / 256, 256, 0, stream>>>(rows, cols, vals, row_start,
                                                      cursor, pairs, nnz);

  // 2) Chebyshev recurrence: init writes T1 -> bufB and acc fully.
  const int sgrid = (N + WGROWS - 1) / WGROWS;
  cheb_spmm_step<0><<<sgrid, 256, 0, stream>>>(pairs, row_start, X, X, bufB, acc,
                                               (float)c[0], (float)c[1], N);

  // Two rotating buffers; from k>=3 T_k legally overwrites T_{k-2}'s buffer
  // (each row reads only its own t0 element before storing it).
  const float* t0p = X;
  const float* t1p = bufB;
  for (int k = 2; k < CHEB_M; ++k) {
    float* dst = (k == 2) ? bufA : const_cast<float*>(t0p);
    cheb_spmm_step<1><<<sgrid, 256, 0, stream>>>(pairs, row_start, t1p, t0p, dst, acc,
                                                 (float)c[k], 0.f, N);
    t0p = t1p;
    t1p = dst;
  }
}